// LeakyIntegrator_49469433315718
// MI455X (gfx1250) — compile-verified
//
#include <hip/hip_runtime.h>

// Problem constants (from reference)
#define LAM_F 0.2f
#define T_   64
#define C_   64
#define H_   128
#define W_   128
#define HW_  (H_ * W_)        // 16384
#define CHW_ (C_ * HW_)       // 1048576

typedef float v4f __attribute__((ext_vector_type(4)));

// ---- CDNA5 async global->LDS path (guarded; falls back to direct loads) ----
#if defined(__AMDGCN__) && __has_builtin(__builtin_amdgcn_global_load_async_to_lds_b128)
#define USE_ASYNC 1
// Builtin prototype (from hipcc diagnostic): pointee is int4 (vector_size(16)),
// src in address_space(1) (global / "__device__"), dst in address_space(3) (LDS).
typedef int v4i_vs __attribute__((vector_size(16)));
typedef __attribute__((address_space(1))) v4i_vs gas_v4i_t;
typedef __attribute__((address_space(3))) v4i_vs las_v4i_t;
__device__ __forceinline__ void stage_b128(const void* g, void* l) {
  // per-lane: copies 16 bytes from global address g to LDS address l, tracked by ASYNCcnt
  __builtin_amdgcn_global_load_async_to_lds_b128((gas_v4i_t*)g, (las_v4i_t*)l,
                                                 /*offset=*/0, /*cpol=*/0);
}
#if __has_builtin(__builtin_amdgcn_s_wait_asynccnt)
#define WAIT_ASYNC(n) __builtin_amdgcn_s_wait_asynccnt(n)
#else
#define WAIT_ASYNC(n) asm volatile("s_wait_asynccnt " #n)
#endif
#else
#define USE_ASYNC 0
#endif

__global__ __launch_bounds__(256)
void leaky_integrator_scan_kernel(const float* __restrict__ img,      // [T,C,H,W]
                                  const float* __restrict__ cov,      // [T,1,H,W]
                                  const unsigned char* __restrict__ add_mask,   // [T] bool
                                  const unsigned char* __restrict__ reset_mask, // [T] bool
                                  float* __restrict__ out_maps,       // [T,C,H,W]
                                  float* __restrict__ out_covs)       // [T,1,H,W]
{
  const int tid = threadIdx.x;
  const int c   = blockIdx.x >> 4;                        // 64 channels
  const int hw  = ((blockIdx.x & 15) << 10) + (tid << 2); // 16 chunks of 1024 per channel
  const size_t base = (size_t)c * HW_ + hw;

  __shared__ float covbuf[2][1024];                       // double-buffered cov chunk (2x4KB)

#if USE_ASYNC
  // prologue: stage cov row chunk for t=0
  stage_b128(cov + hw, &covbuf[0][tid << 2]);
#endif

  // Pack masks into per-thread bitfields (tiny, fully cached; step 0 forced reset).
  unsigned long long rb = 1ull, ab = 0ull;
  {
    const unsigned long long* rq = (const unsigned long long*)reset_mask;
    const unsigned long long* aq = (const unsigned long long*)add_mask;
#pragma unroll
    for (int i = 0; i < 8; ++i) {
      const unsigned long long rv = rq[i];
      const unsigned long long av = aq[i];
#pragma unroll
      for (int b = 0; b < 8; ++b) {
        const int t = i * 8 + b;
        if ((rv >> (8 * b)) & 0xFFull) rb |= (1ull << t);
        if ((av >> (8 * b)) & 0xFFull) ab |= (1ull << t);
      }
    }
  }

  v4f m  = {0.f, 0.f, 0.f, 0.f};   // map carry
  v4f cc = {0.f, 0.f, 0.f, 0.f};   // coverage carry (used only when c==0)
  const bool do_cov = (c == 0);

  for (int t = 0; t < T_; ++t) {
#if USE_ASYNC
    if (t + 1 < T_) {
      // prefetch next timestep's cov chunk, then wait for the current one
      stage_b128(cov + (size_t)(t + 1) * HW_ + hw, &covbuf[(t + 1) & 1][tid << 2]);
      WAIT_ASYNC(1);
    } else {
      WAIT_ASYNC(0);
    }
    const v4f cv = *(const v4f*)&covbuf[t & 1][tid << 2];  // ds_load_b128, own lane's data
#else
    const v4f cv = *(const v4f*)(cov + (size_t)t * HW_ + hw);
#endif

    // streamed read-once input: non-temporal b128 load
    const v4f iv = __builtin_nontemporal_load((const v4f*)(img + (size_t)t * CHW_ + base));

    const bool r = (rb >> t) & 1ull;
    const bool a = (ab >> t) & 1ull;

    // m_upd = (1-l)m + l*img*cov + l*m*(1-cov)  ==  m + (l*cov)*(img - m)
    const v4f mu = m + (cv * LAM_F) * (iv - m);
    m = r ? iv : (a ? mu : m);

    // streamed write-once output: non-temporal b128 store
    __builtin_nontemporal_store(m, (v4f*)(out_maps + (size_t)t * CHW_ + base));

    if (do_cov) {
      const v4f s = cc + cv;
      v4f cu;
      cu.x = fminf(fmaxf(s.x, 0.f), 1.f);
      cu.y = fminf(fmaxf(s.y, 0.f), 1.f);
      cu.z = fminf(fmaxf(s.z, 0.f), 1.f);
      cu.w = fminf(fmaxf(s.w, 0.f), 1.f);
      cc = r ? cv : (a ? cu : cc);
      __builtin_nontemporal_store(cc, (v4f*)(out_covs + (size_t)t * HW_ + hw));
    }
  }
}

extern "C" void kernel_launch(void* const* d_in, const int* in_sizes, int n_in,
                              void* d_out, int out_size, void* d_ws, size_t ws_size,
                              hipStream_t stream) {
  (void)in_sizes; (void)n_in; (void)out_size; (void)d_ws; (void)ws_size;
  // setup_inputs() order: images_w, coverages_w, add_mask, reset_mask
  const float*         img   = (const float*)d_in[0];
  const float*         cov   = (const float*)d_in[1];
  const unsigned char* amask = (const unsigned char*)d_in[2];  // numpy/jax bool = 1 byte
  const unsigned char* rmask = (const unsigned char*)d_in[3];

  float* out_maps = (float*)d_out;                       // [T,C,H,W]
  float* out_covs = out_maps + (size_t)T_ * CHW_;        // [T,1,H,W] follows flat

  // 64 channels x 16 chunks = 1024 blocks; 256 threads = 8 wave32 per block.
  leaky_integrator_scan_kernel<<<dim3(C_ * 16), dim3(256), 0, stream>>>(
      img, cov, amask, rmask, out_maps, out_covs);
}